// MultiBoxLoss_56178172231690
// MI455X (gfx1250) — compile-verified
//
#include <hip/hip_runtime.h>
#include <stdint.h>

#define BB 8
#define PP 100000
#define GG 16
#define NC 41
#define POS_TH 0.5f
#define NEG_TH 0.4f
#define NEGPOS_RATIO 3
#define BBOX_ALPHA 1.5f
#define CONF_ALPHA 1.0f
#define VAR_CXCY 0.1f
#define VAR_WH 0.2f

typedef __attribute__((ext_vector_type(2))) float v2f;
typedef __attribute__((ext_vector_type(8))) float v8f;

__device__ __forceinline__ uint32_t f2ord(float x) {
  uint32_t u = __float_as_uint(x);
  return (u & 0x80000000u) ? ~u : (u | 0x80000000u);
}

// ---------------------------------------------------------------- init
__global__ void k_init(uint64_t* bpkey, int* npos, float* locS, float* cetot) {
  int t = threadIdx.x;
  if (t < BB * GG) bpkey[t] = 0ull;
  if (t < BB) { npos[t] = 0; locS[t] = 0.f; }
  if (t == 0) cetot[0] = 0.f;
}

// ------------------------------------------------------- match (IoU G x P)
__global__ void k_match(const float* __restrict__ priors, const float* __restrict__ gtb,
                        float* __restrict__ btov, int* __restrict__ btidx,
                        uint64_t* __restrict__ bpkey) {
  __shared__ float sg[GG][4];
  __shared__ unsigned long long sbest[GG];
  const int nb = (PP + 255) / 256;
  int b = blockIdx.x / nb;
  int p = (blockIdx.x % nb) * 256 + threadIdx.x;
  if (threadIdx.x < GG * 4) ((float*)sg)[threadIdx.x] = gtb[b * GG * 4 + threadIdx.x];
  if (threadIdx.x < GG) sbest[threadIdx.x] = 0ull;
  __syncthreads();
  if (p < PP) {
    float cx = priors[p * 4 + 0], cy = priors[p * 4 + 1];
    float w = priors[p * 4 + 2], h = priors[p * 4 + 3];
    float px1 = cx - w * 0.5f, py1 = cy - h * 0.5f;
    float px2 = cx + w * 0.5f, py2 = cy + h * 0.5f;
    float areaP = w * h;
    float bestOv = -1.f; int bestG = 0;
    uint32_t ip = ~(uint32_t)p;
#pragma unroll
    for (int g = 0; g < GG; ++g) {
      float gx1 = sg[g][0], gy1 = sg[g][1], gx2 = sg[g][2], gy2 = sg[g][3];
      float ix = fmaxf(fminf(px2, gx2) - fmaxf(px1, gx1), 0.f);
      float iy = fmaxf(fminf(py2, gy2) - fmaxf(py1, gy1), 0.f);
      float inter = ix * iy;
      float areaG = (gx2 - gx1) * (gy2 - gy1);
      float iou = inter / (areaG + areaP - inter);
      if (iou > bestOv) { bestOv = iou; bestG = g; }   // first-occurrence argmax
      unsigned long long key = ((unsigned long long)f2ord(iou) << 32) | ip;
      atomicMax(&sbest[g], key);
    }
    btov[b * PP + p] = bestOv;
    btidx[b * PP + p] = bestG;
  }
  __syncthreads();
  if (threadIdx.x < GG)
    atomicMax((unsigned long long*)&bpkey[b * GG + threadIdx.x], sbest[threadIdx.x]);
}

// ------------------------------------------------ force-match (last g wins)
__global__ void k_force(const uint64_t* __restrict__ bpkey, float* btov, int* btidx) {
  int b = threadIdx.x;
  if (b < BB) {
    for (int g = 0; g < GG; ++g) {
      uint32_t low = (uint32_t)(bpkey[b * GG + g] & 0xffffffffull);
      int p = (int)(~low);
      btov[b * PP + p] = 2.0f;
      btidx[b * PP + p] = g;
    }
  }
}

// ------------------------------------------------------- conf_t + num_pos
__global__ void k_conft(const float* __restrict__ btov, const int* __restrict__ btidx,
                        const int* __restrict__ gtl, int* __restrict__ conf_t,
                        int* __restrict__ npos) {
  __shared__ int scnt;
  const int nb = (PP + 255) / 256;
  int b = blockIdx.x / nb;
  int p = (blockIdx.x % nb) * 256 + threadIdx.x;
  if (threadIdx.x == 0) scnt = 0;
  __syncthreads();
  if (p < PP) {
    float ov = btov[b * PP + p];
    int idx = btidx[b * PP + p];
    int c = gtl[b * GG + idx];
    if (ov < POS_TH) c = -1;
    if (ov < NEG_TH) c = 0;
    conf_t[b * PP + p] = c;
    if (c > 0) atomicAdd(&scnt, 1);
  }
  __syncthreads();
  if (threadIdx.x == 0 && scnt) atomicAdd(&npos[b], scnt);
}

// ------------------------------------------------ localization SmoothL1 sum
__global__ void k_loc(const float* __restrict__ loc, const float* __restrict__ priors,
                      const float* __restrict__ gtb, const int* __restrict__ conf_t,
                      const int* __restrict__ btidx, float* __restrict__ locS) {
  __shared__ float ssum;
  const int nb = (PP + 255) / 256;
  int b = blockIdx.x / nb;
  int p = (blockIdx.x % nb) * 256 + threadIdx.x;
  if (threadIdx.x == 0) ssum = 0.f;
  __syncthreads();
  if (p < PP && conf_t[b * PP + p] > 0) {
    int g = btidx[b * PP + p];
    const float* m = &gtb[(b * GG + g) * 4];
    float pcx = priors[p * 4], pcy = priors[p * 4 + 1];
    float pw = priors[p * 4 + 2], ph = priors[p * 4 + 3];
    float mx1 = m[0], my1 = m[1], mx2 = m[2], my2 = m[3];
    float t4[4];
    t4[0] = ((mx1 + mx2) * 0.5f - pcx) / (VAR_CXCY * pw);
    t4[1] = ((my1 + my2) * 0.5f - pcy) / (VAR_CXCY * ph);
    t4[2] = logf(fmaxf((mx2 - mx1) / pw, 1e-8f)) / VAR_WH;
    t4[3] = logf(fmaxf((my2 - my1) / ph, 1e-8f)) / VAR_WH;
    const float* ld = &loc[(size_t)(b * PP + p) * 4];
    float s = 0.f;
#pragma unroll
    for (int i = 0; i < 4; ++i) {
      float d = ld[i] - t4[i]; float ad = fabsf(d);
      s += (ad < 1.f) ? 0.5f * d * d : ad - 0.5f;
    }
    atomicAdd(&ssum, s);
  }
  __syncthreads();
  if (threadIdx.x == 0 && ssum != 0.f) atomicAdd(&locS[b], ssum);
}

// -------------------- logsumexp via V_WMMA_F32_16X16X4_F32 (16 priors/wave)
__global__ void k_conf(const float* __restrict__ conf, const int* __restrict__ conf_t,
                       float* __restrict__ ce, uint64_t* __restrict__ keys) {
  const int nb = (PP + 127) / 128;
  int b = blockIdx.x / nb;
  int p0 = (blockIdx.x % nb) * 128 + ((int)threadIdx.x >> 5) * 16;
  int l = threadIdx.x & 31;
  int r = l & 15, hh = l >> 4;              // A 16x4 f32 layout: lanes 0-15 K={0,1}, 16-31 K={2,3}
  int p = p0 + r;
  int pc = p < PP ? p : PP - 1;
  const float* row = conf + (size_t)(b * PP + pc) * NC;
  float a0[11], a1[11];
#pragma unroll
  for (int c = 0; c < 11; ++c) {            // 11 chunks of K=4 -> 44 padded classes
    int c0 = 4 * c + 2 * hh;
    a0[c] = (c0 < NC) ? row[c0] : -__builtin_inff();
    a1[c] = (c0 + 1 < NC) ? row[c0 + 1] : -__builtin_inff();
  }
  float m = -__builtin_inff();
#pragma unroll
  for (int c = 0; c < 11; ++c) m = fmaxf(m, fmaxf(a0[c], a1[c]));
  m = fmaxf(m, __shfl_xor(m, 16, 32));      // combine lane halves -> per-row max

  v8f acc = {0.f, 0.f, 0.f, 0.f, 0.f, 0.f, 0.f, 0.f};
  v2f ones; ones.x = 1.f; ones.y = 1.f;     // all-ones B (layout-independent)
#pragma unroll
  for (int c = 0; c < 11; ++c) {
    v2f av; av.x = __expf(a0[c] - m); av.y = __expf(a1[c] - m);
    acc = __builtin_amdgcn_wmma_f32_16x16x4_f32(false, av, false, ones,
                                                (short)0, acc, false, false);
  }
  // D[m][n] = rowsum(m) for every n; row j in lane0 c[j] (j<8), lane16 c[j] (rows 8..15)
  float rs = 0.f;
#pragma unroll
  for (int j = 0; j < 8; ++j) {
    float slo = __shfl(acc[j], 0, 32);
    float shi = __shfl(acc[j], 16, 32);
    if (r == j) rs = slo;
    if (r == j + 8) rs = shi;
  }
  float lse = m + __logf(rs);

  if (hh == 0 && p < PP) {
    size_t o = (size_t)b * PP + p;
    int ct = conf_t[o];
    int lab = ct > 0 ? ct : 0;
    float c0v = row[0];
    float clv = row[lab];
    ce[o] = lse - clv;                                       // -log_softmax[label]
    bool pos = ct > 0;
    float lc = (pos || ct < 0) ? 0.f : (lse - c0v);          // OHEM metric (masked)
    keys[o] = ((uint64_t)f2ord(lc) << 32) | (uint32_t)(~(uint32_t)p);  // unique key
  }
}

// ------------------------------------------------------- radix-select setup
__global__ void k_selinit(const int* npos, uint64_t* prefix, int* kk, int* nneg) {
  int b = threadIdx.x;
  if (b < BB) {
    int nn = NEGPOS_RATIO * npos[b];
    if (nn > PP - 1) nn = PP - 1;
    nneg[b] = nn; kk[b] = nn; prefix[b] = 0ull;
  }
}

__global__ void k_histzero(uint32_t* hist) {
  int i = blockIdx.x * 256 + threadIdx.x;
  if (i < BB * 256) hist[i] = 0;
}

__global__ void k_hist(const uint64_t* __restrict__ keys, const uint64_t* __restrict__ prefix,
                       const int* __restrict__ kk, uint32_t* __restrict__ hist,
                       int shift, uint64_t himask) {
  __shared__ uint32_t sh[256];
  const int nb = (PP + 255) / 256;
  int b = blockIdx.x / nb;
  int p = (blockIdx.x % nb) * 256 + threadIdx.x;
  sh[threadIdx.x] = 0;
  __syncthreads();
  if (kk[b] > 0 && p < PP) {
    uint64_t k = keys[(size_t)b * PP + p];
    if (((k ^ prefix[b]) & himask) == 0)
      atomicAdd(&sh[(uint32_t)(k >> shift) & 255u], 1u);
  }
  __syncthreads();
  if (sh[threadIdx.x]) atomicAdd(&hist[b * 256 + threadIdx.x], sh[threadIdx.x]);
}

__global__ void k_scan(const uint32_t* __restrict__ hist, uint64_t* prefix, int* kk, int shift) {
  int b = threadIdx.x;
  if (b < BB && kk[b] > 0) {
    int target = kk[b];
    unsigned cum = 0;
    for (int bin = 255; bin >= 0; --bin) {
      unsigned c = hist[b * 256 + bin];
      if (cum + c >= (unsigned)target) {
        prefix[b] |= ((uint64_t)(unsigned)bin) << shift;
        kk[b] = target - (int)cum;
        break;
      }
      cum += c;
    }
  }
}

// ------------------------------------------------------- masked CE reduction
__global__ void k_final(const int* __restrict__ conf_t, const float* __restrict__ ce,
                        const uint64_t* __restrict__ keys, const uint64_t* __restrict__ prefix,
                        const int* __restrict__ nneg, float* __restrict__ cetot) {
  __shared__ float ssum;
  const int nb = (PP + 255) / 256;
  int b = blockIdx.x / nb;
  int p = (blockIdx.x % nb) * 256 + threadIdx.x;
  if (threadIdx.x == 0) ssum = 0.f;
  __syncthreads();
  if (p < PP) {
    size_t o = (size_t)b * PP + p;
    int ct = conf_t[o];
    bool pos = ct > 0;
    bool neg = (nneg[b] > 0) && !pos && ct >= 0 && (keys[o] >= prefix[b]);
    if (pos || neg) atomicAdd(&ssum, ce[o]);
  }
  __syncthreads();
  if (threadIdx.x == 0 && ssum != 0.f) atomicAdd(cetot, ssum);
}

__global__ void k_fin(const int* npos, const float* locS, const float* cetot, float* out) {
  if (threadIdx.x == 0) {
    float lb = 0.f; int tp = 0;
    for (int b = 0; b < BB; ++b) {
      int np = npos[b]; tp += np;
      int d = np > 1 ? np : 1;
      lb += locS[b] / (float)d;
    }
    lb *= BBOX_ALPHA;
    int tpc = tp > 1 ? tp : 1;
    float lc = cetot[0] * CONF_ALPHA / (float)tpc;
    out[0] = (lb + lc) / (float)BB;
  }
}

// ---------------------------------------------------------------- launcher
extern "C" void kernel_launch(void* const* d_in, const int* in_sizes, int n_in,
                              void* d_out, int out_size, void* d_ws, size_t ws_size,
                              hipStream_t stream) {
  const float* loc    = (const float*)d_in[0];
  const float* conf   = (const float*)d_in[1];
  const float* priors = (const float*)d_in[2];
  const float* gtb    = (const float*)d_in[3];
  const int*   gtl    = (const int*)d_in[4];
  float* out = (float*)d_out;

  const size_t BP = (size_t)BB * PP;
  char* base = (char*)d_ws;
  size_t off = 0;
  auto carve = [&](size_t bytes) -> char* {
    char* pt = base + off;
    off = (off + bytes + 255) & ~(size_t)255;
    return pt;
  };
  int*      conf_t = (int*)carve(BP * 4);
  int*      btidx  = (int*)carve(BP * 4);
  float*    btov   = (float*)carve(BP * 4);
  float*    ce     = (float*)carve(BP * 4);
  uint64_t* keys   = (uint64_t*)carve(BP * 8);
  uint64_t* bpkey  = (uint64_t*)carve((size_t)BB * GG * 8);
  uint32_t* hist   = (uint32_t*)carve((size_t)BB * 256 * 4);
  uint64_t* prefix = (uint64_t*)carve(BB * 8);
  int*      kk     = (int*)carve(BB * 4);
  int*      nneg   = (int*)carve(BB * 4);
  int*      npos   = (int*)carve(BB * 4);
  float*    locS   = (float*)carve(BB * 4);
  float*    cetot  = (float*)carve(4);

  const int nb1 = (PP + 255) / 256;   // 391
  const int nb2 = (PP + 127) / 128;   // 782

  k_init<<<1, 128, 0, stream>>>(bpkey, npos, locS, cetot);
  k_match<<<BB * nb1, 256, 0, stream>>>(priors, gtb, btov, btidx, bpkey);
  k_force<<<1, 32, 0, stream>>>(bpkey, btov, btidx);
  k_conft<<<BB * nb1, 256, 0, stream>>>(btov, btidx, gtl, conf_t, npos);
  k_loc<<<BB * nb1, 256, 0, stream>>>(loc, priors, gtb, conf_t, btidx, locS);
  k_conf<<<BB * nb2, 256, 0, stream>>>(conf, conf_t, ce, keys);
  k_selinit<<<1, 32, 0, stream>>>(npos, prefix, kk, nneg);
  for (int shift = 56; shift >= 0; shift -= 8) {
    uint64_t himask = (shift + 8 >= 64) ? 0ull : (~0ull << (shift + 8));
    k_histzero<<<8, 256, 0, stream>>>(hist);
    k_hist<<<BB * nb1, 256, 0, stream>>>(keys, prefix, kk, hist, shift, himask);
    k_scan<<<1, 32, 0, stream>>>(hist, prefix, kk, shift);
  }
  k_final<<<BB * nb1, 256, 0, stream>>>(conf_t, ce, keys, prefix, nneg, cetot);
  k_fin<<<1, 32, 0, stream>>>(npos, locS, cetot, out);
}